// QueryPooling_58308476011171
// MI455X (gfx1250) — compile-verified
//
#include <hip/hip_runtime.h>
#include <hip/hip_bf16.h>

// Fused query-pooled cross-attention for MI455X (gfx1250), wave32.
// One workgroup per graph; flash-attention streaming over node chunks.
// - All GEMMs use V_WMMA_F32_16X16X4_F32 (full fp32; problem is HBM-bound
//   at ~11us so fp32 WMMA throughput is ample and numerics match reference).
// - X chunks are streamed with GLOBAL_LOAD_ASYNC_TO_LDS_B128 (ASYNCcnt) into
//   a double buffer so HBM latency overlaps the WMMA work.

#define B_GRAPHS 128
#define N2       32
#define FDIM     256
#define HEADS    4
#define DHEAD    64
#define TOTAL_N  262144
#define CCHUNK   32     // nodes per streamed chunk
#define FP       260    // padded row of FDIM floats (16B-aligned stride)
#define SP       36     // padded row of CCHUNK floats

typedef float v2f __attribute__((ext_vector_type(2)));
typedef float v8f __attribute__((ext_vector_type(8)));

__device__ __forceinline__ v8f vzero8() {
  v8f z = {0.f,0.f,0.f,0.f,0.f,0.f,0.f,0.f};
  return z;
}

// D = A(16x4,f32) * B(4x16,f32) + C(16x16,f32)
__device__ __forceinline__ v8f wmma4(v2f a, v2f b, v8f c) {
  return __builtin_amdgcn_wmma_f32_16x16x4_f32(
      /*neg_a=*/false, a, /*neg_b=*/false, b,
      /*c_mod=*/(short)0, c, /*reuse_a=*/false, /*reuse_b=*/false);
}

// Async global->LDS copy, 16 bytes per lane.  lds_off = byte offset within the
// workgroup LDS allocation (low 32 bits of the flat shared address).
__device__ __forceinline__ void async_ld_b128(unsigned lds_off, const void* gaddr) {
  asm volatile("global_load_async_to_lds_b128 %0, %1, off"
               :: "v"(lds_off), "v"(gaddr) : "memory");
}
__device__ __forceinline__ void wait_async0() {
  asm volatile("s_wait_asynccnt 0x0" ::: "memory");
}

// ---------------------------------------------------------------------------
// Kernel A: graph start offsets from sorted batch array.
// ---------------------------------------------------------------------------
__global__ void graph_offsets_kernel(const int* __restrict__ batch,
                                     int* __restrict__ starts) {
  int i = blockIdx.x * blockDim.x + threadIdx.x;
  if (i >= TOTAL_N) return;
  int b = batch[i];
  if (i == 0 || batch[i - 1] != b) starts[b] = i;
  if (i == TOTAL_N - 1) starts[B_GRAPHS] = TOTAL_N;
}

// ---------------------------------------------------------------------------
// Kernel B: q = (queries @ Wq^T + bq) * (1/sqrt(d)).  32x256, trivial.
// ---------------------------------------------------------------------------
__global__ void q_proj_kernel(const float* __restrict__ queries,
                              const float* __restrict__ Wq,
                              const float* __restrict__ bq,
                              float* __restrict__ qs) {
  const int qr  = blockIdx.x;    // 0..31
  const int col = threadIdx.x;   // 0..255
  const float* qv = queries + qr * FDIM;
  const float* wr = Wq + (size_t)col * FDIM;
  float s = 0.f;
  for (int k = 0; k < FDIM; ++k) s = fmaf(qv[k], wr[k], s);
  qs[qr * FDIM + col] = (s + bq[col]) * 0.125f;   // 1/sqrt(64)
}

// ---------------------------------------------------------------------------
// Kernel C: fused K/V projection + streaming softmax attention + out-proj.
// ---------------------------------------------------------------------------
__global__ __launch_bounds__(256)
void fused_graph_attn_kernel(const float* __restrict__ x,
                             const int*   __restrict__ starts,
                             const float* __restrict__ qs,   // pre-scaled Q [32][256]
                             const float* __restrict__ Wk,   // [256][256] row-major
                             const float* __restrict__ Wv,   // [256][256] row-major
                             const float* __restrict__ bv,   // [256]
                             const float* __restrict__ Wo,   // [256][256] row-major
                             const float* __restrict__ bo,   // [256]
                             float* __restrict__ out) {
  __shared__ __align__(16) float xc2[2][CCHUNK][FP];  // double-buffered node features
  __shared__ __align__(16) float kc [CCHUNK][FP];     // projected keys
  __shared__ __align__(16) float vcs[CCHUNK][FP];     // projected values
  __shared__ __align__(16) float ql [N2][FP];         // scaled queries
  __shared__ __align__(16) float ol [N2][FP];         // pooled output before out-proj
  __shared__ __align__(16) float sl [HEADS][N2][SP];  // scores -> probabilities
  __shared__ float ml [HEADS * N2];                   // running max
  __shared__ float ll [HEADS * N2];                   // running denominator
  __shared__ float rl [HEADS * N2];                   // per-chunk rescale factor

  const int tid  = threadIdx.x;
  const int w    = tid >> 5;         // wave id 0..7
  const int lane = tid & 31;
  const int l16  = lane & 15;
  const int hi   = lane >> 4;        // lane half
  const int koff = 2 * hi;           // K-offset inside a 16x4 fragment

  // stage queries into LDS; init softmax state
  for (int i = tid; i < N2 * FDIM; i += 256) ql[i >> 8][i & 255] = qs[i];
  if (tid < HEADS * N2) { ml[tid] = -1e30f; ll[tid] = 0.f; }

  // persistent O accumulators: wave owns head hO, d-cols [ntO0*16, ntO0*16+32)
  const int hO   = w >> 1;
  const int ntO0 = (w & 1) * 2;
  v8f oacc[2][2];
  #pragma unroll
  for (int mt = 0; mt < 2; ++mt)
    #pragma unroll
    for (int j = 0; j < 2; ++j) oacc[mt][j] = vzero8();

  const int g   = blockIdx.x;
  const int s0  = starts[g];
  const int s1  = starts[g + 1];
  const int nch = (s1 - s0 + CCHUNK - 1) / CCHUNK;

  // ---- async chunk loader: 2048 x b128 per chunk, 8 per thread ----
  auto load_chunk_async = [&](int buf, int base, int valid) {
    for (int i = tid; i < CCHUNK * (FDIM / 4); i += 256) {
      const int r  = i >> 6;             // 64 x b128 per row
      const int c4 = (i & 63) * 4;
      float* lp = &xc2[buf][r][c4];
      if (r < valid) {
        async_ld_b128((unsigned)(uintptr_t)lp,
                      (const void*)(x + (size_t)(base + r) * FDIM + c4));
      } else {
        lp[0] = 0.f; lp[1] = 0.f; lp[2] = 0.f; lp[3] = 0.f;  // tail zero-fill
      }
    }
  };

  // preload chunk 0
  load_chunk_async(0, s0, min(CCHUNK, s1 - s0));

  for (int c = 0; c < nch; ++c) {
    const int cur   = c & 1;
    const int base  = s0 + c * CCHUNK;
    const int valid = min(CCHUNK, s1 - base);

    wait_async0();        // our slice of chunk `c` has landed in LDS
    __syncthreads();      // everyone's slice visible; prev iter's LDS reads done

    if (c + 1 < nch) {    // overlap next chunk's HBM traffic with compute
      const int nb = s0 + (c + 1) * CCHUNK;
      load_chunk_async(cur ^ 1, nb, min(CCHUNK, s1 - nb));
      if (c + 2 < nch)    // and keep L2 one chunk ahead
        __builtin_prefetch(x + (size_t)(s0 + (c + 2) * CCHUNK) * FDIM + tid * 32, 0, 0);
    }

    const float (*xc)[FP] = xc2[cur];

    // ---- GEMM1: Kc = Xc @ Wk^T, Vc = Xc @ Wv^T  (waves 0-3: K, 4-7: V) ----
    {
      const int   proj = w >> 2;
      const int   nt0  = (w & 3) * 4;
      const float* Wp  = proj ? Wv : Wk;
      float*       dst = proj ? &vcs[0][0] : &kc[0][0];
      v8f acc[2][4];
      #pragma unroll
      for (int mt = 0; mt < 2; ++mt)
        #pragma unroll
        for (int j = 0; j < 4; ++j) acc[mt][j] = vzero8();

      for (int k0 = 0; k0 < FDIM; k0 += 4) {
        v2f a0, a1, b[4];
        a0.x = xc[l16][k0 + koff];      a0.y = xc[l16][k0 + koff + 1];
        a1.x = xc[16 + l16][k0 + koff]; a1.y = xc[16 + l16][k0 + koff + 1];
        #pragma unroll
        for (int j = 0; j < 4; ++j) {
          const float* p = Wp + (size_t)((nt0 + j) * 16 + l16) * FDIM + k0 + koff;
          b[j].x = p[0]; b[j].y = p[1];
        }
        #pragma unroll
        for (int j = 0; j < 4; ++j) {
          acc[0][j] = wmma4(a0, b[j], acc[0][j]);
          acc[1][j] = wmma4(a1, b[j], acc[1][j]);
        }
      }
      #pragma unroll
      for (int mt = 0; mt < 2; ++mt)
        #pragma unroll
        for (int j = 0; j < 4; ++j)
          #pragma unroll
          for (int r = 0; r < 8; ++r)
            dst[(mt * 16 + hi * 8 + r) * FP + (nt0 + j) * 16 + l16] = acc[mt][j][r];
    }
    __syncthreads();

    // ---- GEMM2: S_h = Q_h @ Kc_h^T  (wave -> head w>>1, node-tile w&1) ----
    {
      const int h  = w >> 1;
      const int nt = w & 1;
      v8f sacc[2] = {vzero8(), vzero8()};
      for (int k0 = 0; k0 < DHEAD; k0 += 4) {
        const int kk = h * DHEAD + k0 + koff;
        v2f a0, a1, b;
        a0.x = ql[l16][kk];            a0.y = ql[l16][kk + 1];
        a1.x = ql[16 + l16][kk];       a1.y = ql[16 + l16][kk + 1];
        b.x  = kc[nt * 16 + l16][kk];  b.y  = kc[nt * 16 + l16][kk + 1];
        sacc[0] = wmma4(a0, b, sacc[0]);
        sacc[1] = wmma4(a1, b, sacc[1]);
      }
      #pragma unroll
      for (int mt = 0; mt < 2; ++mt)
        #pragma unroll
        for (int r = 0; r < 8; ++r)
          sl[h][mt * 16 + hi * 8 + r][nt * 16 + l16] = sacc[mt][r];
    }
    __syncthreads();

    // ---- online softmax over this chunk (128 rows = head*query) ----
    if (tid < HEADS * N2) {
      const int h = tid >> 5, q = tid & 31;
      float mo = ml[tid];
      float cm = -1e30f;
      for (int n = 0; n < valid; ++n) cm = fmaxf(cm, sl[h][q][n]);
      const float mn = fmaxf(mo, cm);
      const float sc = __expf(mo - mn);
      float sum = 0.f;
      for (int n = 0; n < CCHUNK; ++n) {
        const float p = (n < valid) ? __expf(sl[h][q][n] - mn) : 0.f;
        sl[h][q][n] = p;
        sum += p;
      }
      ml[tid] = mn;
      ll[tid] = ll[tid] * sc + sum;
      rl[tid] = sc;
    }
    __syncthreads();

    // ---- GEMM3: O = O*rescale + P @ Vc_h  (persistent accumulators) ----
    {
      #pragma unroll
      for (int mt = 0; mt < 2; ++mt)
        #pragma unroll
        for (int r = 0; r < 8; ++r) {
          const float sc = rl[hO * N2 + mt * 16 + hi * 8 + r];
          #pragma unroll
          for (int j = 0; j < 2; ++j) oacc[mt][j][r] *= sc;
        }
      for (int k0 = 0; k0 < CCHUNK; k0 += 4) {
        const int kk = k0 + koff;
        v2f a0, a1, b[2];
        a0.x = sl[hO][l16][kk];       a0.y = sl[hO][l16][kk + 1];
        a1.x = sl[hO][16 + l16][kk];  a1.y = sl[hO][16 + l16][kk + 1];
        #pragma unroll
        for (int j = 0; j < 2; ++j) {
          const int col = hO * DHEAD + (ntO0 + j) * 16 + l16;
          b[j].x = vcs[kk][col];
          b[j].y = vcs[kk + 1][col];
        }
        #pragma unroll
        for (int j = 0; j < 2; ++j) {
          oacc[0][j] = wmma4(a0, b[j], oacc[0][j]);
          oacc[1][j] = wmma4(a1, b[j], oacc[1][j]);
        }
      }
    }
  } // chunk loop

  // ---- finalize: normalize, add V bias (sum(attn)==1), stage O ----
  __syncthreads();
  #pragma unroll
  for (int mt = 0; mt < 2; ++mt)
    #pragma unroll
    for (int j = 0; j < 2; ++j)
      #pragma unroll
      for (int r = 0; r < 8; ++r) {
        const int q   = mt * 16 + hi * 8 + r;
        const int col = hO * DHEAD + (ntO0 + j) * 16 + l16;
        ol[q][col] = oacc[mt][j][r] / ll[hO * N2 + q] + bv[col];
      }
  __syncthreads();

  // ---- GEMM4: out = O @ Wo^T + bo ----
  {
    const int nt0 = w * 2;   // 2 of 16 n-tiles per wave
    v8f f[2][2];
    #pragma unroll
    for (int mt = 0; mt < 2; ++mt)
      #pragma unroll
      for (int j = 0; j < 2; ++j) f[mt][j] = vzero8();

    for (int k0 = 0; k0 < FDIM; k0 += 4) {
      v2f a0, a1, b[2];
      a0.x = ol[l16][k0 + koff];      a0.y = ol[l16][k0 + koff + 1];
      a1.x = ol[16 + l16][k0 + koff]; a1.y = ol[16 + l16][k0 + koff + 1];
      #pragma unroll
      for (int j = 0; j < 2; ++j) {
        const float* p = Wo + (size_t)((nt0 + j) * 16 + l16) * FDIM + k0 + koff;
        b[j].x = p[0]; b[j].y = p[1];
      }
      #pragma unroll
      for (int j = 0; j < 2; ++j) {
        f[0][j] = wmma4(a0, b[j], f[0][j]);
        f[1][j] = wmma4(a1, b[j], f[1][j]);
      }
    }
    #pragma unroll
    for (int mt = 0; mt < 2; ++mt)
      #pragma unroll
      for (int j = 0; j < 2; ++j)
        #pragma unroll
        for (int r = 0; r < 8; ++r) {
          const int q   = mt * 16 + hi * 8 + r;
          const int col = (nt0 + j) * 16 + l16;
          out[((size_t)g * N2 + q) * FDIM + col] = f[mt][j][r] + bo[col];
        }
  }
}

// ---------------------------------------------------------------------------
extern "C" void kernel_launch(void* const* d_in, const int* in_sizes, int n_in,
                              void* d_out, int out_size, void* d_ws, size_t ws_size,
                              hipStream_t stream) {
  (void)in_sizes; (void)n_in; (void)out_size; (void)ws_size;
  const float* x       = (const float*)d_in[0];
  const int*   batch   = (const int*)  d_in[1];
  const float* queries = (const float*)d_in[2];
  const float* Win     = (const float*)d_in[3];   // [768][256]
  const float* bin     = (const float*)d_in[4];   // [768]
  const float* Wo      = (const float*)d_in[5];   // [256][256]
  const float* bo      = (const float*)d_in[6];   // [256]
  float*       out     = (float*)d_out;

  float* qs     = (float*)d_ws;                                            // 32*256 f32
  int*   starts = (int*)((char*)d_ws + (size_t)N2 * FDIM * sizeof(float)); // 129 i32

  graph_offsets_kernel<<<TOTAL_N / 256, 256, 0, stream>>>(batch, starts);
  q_proj_kernel<<<N2, FDIM, 0, stream>>>(queries, Win /*Wq*/, bin /*bq*/, qs);
  fused_graph_attn_kernel<<<B_GRAPHS, 256, 0, stream>>>(
      x, starts, qs,
      Win + (size_t)FDIM * FDIM,       // Wk
      Win + (size_t)2 * FDIM * FDIM,   // Wv
      bin + 2 * FDIM,                  // bv
      Wo, bo, out);
}